// FastSpeech2VarianceAdaptor_80135499809293
// MI455X (gfx1250) — compile-verified
//
#include <hip/hip_runtime.h>
#include <hip/hip_bf16.h>

#define BATCH  32
#define SEQ    2048
#define DIM    256          // D == H == 256
#define MAXOUT 6144         // (DUR_MAX-1)*S
#define NBINS  255
#define WPACK  (3 * 64 * 2 * 256 * 2)   // packed weight elems per conv = 196608

typedef float v2f __attribute__((ext_vector_type(2)));
typedef float v8f __attribute__((ext_vector_type(8)));

// ---------------------------------------------------------------------------
// Pack conv weight (3,256,256)[k][d][h] into WMMA-fragment order:
//   wp[t][q][khalf][n][j] = w[t][4q + 2*khalf + j][n]
// so each lane's B K-pair is a contiguous float2 (one global_load_b64),
// coalesced 128B per half-wave.
// ---------------------------------------------------------------------------
__global__ __launch_bounds__(256) void pack_w_kernel(
    const float* __restrict__ w, float* __restrict__ wp)
{
    const int o = blockIdx.x * 256 + threadIdx.x;   // 0 .. WPACK-1
    const int j = o & 1;
    const int n = (o >> 1) & 255;
    const int h = (o >> 9) & 1;
    const int q = (o >> 10) & 63;
    const int t = o >> 16;
    wp[o] = w[t * (DIM * DIM) + (4 * q + 2 * h + j) * DIM + n];
}

// ---------------------------------------------------------------------------
// Fused conv1d(K=3,pad=1) -> +bias -> ReLU -> LayerNorm over channels.
// Block = 256 threads (8 waves). Each block computes 32 consecutive (b,s)
// rows x all 256 output channels. Each wave: 2 row tiles x 2 col tiles
// (4 v8f accumulators) so each B fragment feeds two WMMAs
// (0.5 global_load_b64 per WMMA). GEMM via V_WMMA_F32_16X16X4_F32 over
// Ktot = 3*256 = 768 (fp32 exact: outputs feed searchsorted w/ 0.031 bins).
// ---------------------------------------------------------------------------
__global__ __launch_bounds__(256) void conv_ln_kernel(
    const float* __restrict__ x,      // (B,S,256)
    const float* __restrict__ wp,     // packed weights, WPACK floats
    const float* __restrict__ bias,   // (256)
    const float* __restrict__ gamma,  // (256)
    const float* __restrict__ beta,   // (256)
    float* __restrict__ out)          // (B,S,256)
{
    __shared__ float xs[34 * DIM];    // rows s0-1 .. s0+32 (halo for K=3)
    __shared__ float zs[32 * DIM];    // post-ReLU tile for LayerNorm

    const int tile = blockIdx.x;                  // 0 .. B*S/32 - 1
    const int s0   = (tile * 32) % SEQ;
    const int b    = (tile * 32) / SEQ;
    const float* xb = x + (size_t)b * SEQ * DIM;

    const int tid = threadIdx.x;

    // Cooperative load of 34 input rows (zero padding outside [0,SEQ))
    for (int i = tid; i < 34 * (DIM / 4); i += 256) {
        int r = i / (DIM / 4);
        int s = s0 - 1 + r;
        float4 v = make_float4(0.f, 0.f, 0.f, 0.f);
        if (s >= 0 && s < SEQ)
            v = ((const float4*)(xb + (size_t)s * DIM))[i % (DIM / 4)];
        ((float4*)xs)[i] = v;
    }
    __syncthreads();

    const int wave  = tid >> 5;       // 0..7
    const int lane  = tid & 31;
    const int m     = lane & 15;      // row-in-tile / col-in-tile
    const int khalf = lane >> 4;      // selects K pair within fragment
    const int n0    = wave * 32;      // this wave owns columns n0..n0+31

    v8f acc00 = {};                   // rows 0-15,  cols n0..n0+15
    v8f acc01 = {};                   // rows 0-15,  cols n0+16..n0+31
    v8f acc10 = {};                   // rows 16-31, cols n0..n0+15
    v8f acc11 = {};                   // rows 16-31, cols n0+16..n0+31

    for (int t = 0; t < 3; ++t) {
        const float*  xr0 = xs + (m + t) * DIM;        // A rows 0-15, tap t
        const float*  xr1 = xs + (m + 16 + t) * DIM;   // A rows 16-31, tap t
        const float2* wpt = ((const float2*)wp) + (size_t)t * (64 * 2 * 256);
#pragma unroll 4
        for (int q = 0; q < 64; ++q) {
            const int ka = 4 * q + 2 * khalf;
            v2f a0; a0.x = xr0[ka]; a0.y = xr0[ka + 1];
            v2f a1; a1.x = xr1[ka]; a1.y = xr1[ka + 1];
            const int wb = (q * 2 + khalf) * 256;
            const float2 f0 = wpt[wb + n0 + m];
            const float2 f1 = wpt[wb + n0 + 16 + m];
            v2f b0; b0.x = f0.x; b0.y = f0.y;
            v2f b1; b1.x = f1.x; b1.y = f1.y;
            acc00 = __builtin_amdgcn_wmma_f32_16x16x4_f32(
                        false, a0, false, b0, (short)0, acc00, false, false);
            acc01 = __builtin_amdgcn_wmma_f32_16x16x4_f32(
                        false, a0, false, b1, (short)0, acc01, false, false);
            acc10 = __builtin_amdgcn_wmma_f32_16x16x4_f32(
                        false, a1, false, b0, (short)0, acc10, false, false);
            acc11 = __builtin_amdgcn_wmma_f32_16x16x4_f32(
                        false, a1, false, b1, (short)0, acc11, false, false);
        }
    }

    // bias + ReLU, scatter D-matrix layout into shared tile
    const float bia0 = bias[n0 + m];
    const float bia1 = bias[n0 + 16 + m];
#pragma unroll
    for (int i = 0; i < 8; ++i) {
        int row = i + 8 * khalf;      // C/D layout: VGPR i holds M=i / M=8+i
        zs[row * DIM + n0 + m]             = fmaxf(acc00[i] + bia0, 0.f);
        zs[row * DIM + n0 + 16 + m]        = fmaxf(acc01[i] + bia1, 0.f);
        zs[(row + 16) * DIM + n0 + m]      = fmaxf(acc10[i] + bia0, 0.f);
        zs[(row + 16) * DIM + n0 + 16 + m] = fmaxf(acc11[i] + bia1, 0.f);
    }
    __syncthreads();

    // LayerNorm: each wave normalizes 4 rows (wave32 shuffle reduction)
    for (int r = wave * 4; r < wave * 4 + 4; ++r) {
        float vals[8];
        float sum = 0.f, sumsq = 0.f;
#pragma unroll
        for (int j = 0; j < 8; ++j) {
            float v = zs[r * DIM + lane + 32 * j];
            vals[j] = v; sum += v; sumsq += v * v;
        }
#pragma unroll
        for (int off = 16; off > 0; off >>= 1) {
            sum   += __shfl_xor(sum, off, 32);
            sumsq += __shfl_xor(sumsq, off, 32);
        }
        const float mean = sum * (1.f / DIM);
        const float var  = sumsq * (1.f / DIM) - mean * mean;
        const float rstd = rsqrtf(var + 1e-5f);
        const size_t base = ((size_t)b * SEQ + s0 + r) * DIM;
#pragma unroll
        for (int j = 0; j < 8; ++j) {
            int c = lane + 32 * j;
            out[base + c] = (vals[j] - mean) * rstd * gamma[c] + beta[c];
        }
    }
}

// ---------------------------------------------------------------------------
// Projection H->1, optional searchsorted(bins,left) + embedding add.
// Block = 256 threads = 8 waves, one wave per (b,s) row.
// ---------------------------------------------------------------------------
__global__ __launch_bounds__(256) void proj_embed_kernel(
    const float* __restrict__ h2,       // (B*S,256)
    const float* __restrict__ pw,       // (256)
    const float* __restrict__ pb,       // (1)
    const float* __restrict__ bins,     // (255) or null
    const float* __restrict__ embed,    // (256,256) or null
    const float* __restrict__ addsrc,   // (B*S,256) or null
    float* __restrict__ hacc,           // (B*S,256) or null
    float* __restrict__ scalar_out,     // (B*S)
    int mode)
{
    const int row  = blockIdx.x * 8 + (threadIdx.x >> 5);
    const int lane = threadIdx.x & 31;
    const float* hrow = h2 + (size_t)row * DIM;

    float acc = 0.f;
#pragma unroll
    for (int j = 0; j < 8; ++j) {
        int c = lane + 32 * j;
        acc += hrow[c] * pw[c];
    }
#pragma unroll
    for (int off = 16; off > 0; off >>= 1)
        acc += __shfl_xor(acc, off, 32);
    const float val = acc + pb[0];
    if (lane == 0) scalar_out[row] = val;

    if (mode != 0) {
        // searchsorted(bins, val, 'left') == count(bins[j] < val)
        int cnt = 0;
#pragma unroll
        for (int j = 0; j < 8; ++j) {
            int bi = lane + 32 * j;
            if (bi < NBINS && bins[bi] < val) cnt++;
        }
#pragma unroll
        for (int off = 16; off > 0; off >>= 1)
            cnt += __shfl_xor(cnt, off, 32);
        const float* erow = embed + (size_t)cnt * DIM;
        const float* srow = addsrc + (size_t)row * DIM;
        float*       drow = hacc + (size_t)row * DIM;
#pragma unroll
        for (int j = 0; j < 8; ++j) {
            int c = lane + 32 * j;
            drow[c] = srow[c] + erow[c];
        }
    }
}

// ---------------------------------------------------------------------------
// Per-batch inclusive cumsum of durations; also writes out_lens (as float).
// ---------------------------------------------------------------------------
__global__ __launch_bounds__(256) void cumsum_kernel(
    const int* __restrict__ dur, int* __restrict__ cum,
    float* __restrict__ lens_out)
{
    __shared__ int part[256];
    const int b = blockIdx.x, tid = threadIdx.x;
    const int* d = dur + (size_t)b * SEQ;

    int loc[8];
    int run = 0;
#pragma unroll
    for (int j = 0; j < 8; ++j) { run += d[tid * 8 + j]; loc[j] = run; }
    part[tid] = run;
    __syncthreads();
    if (tid == 0) {
        int a = 0;
        for (int i = 0; i < 256; ++i) { int t = part[i]; part[i] = a; a += t; }
    }
    __syncthreads();
    const int off = part[tid];
#pragma unroll
    for (int j = 0; j < 8; ++j) cum[(size_t)b * SEQ + tid * 8 + j] = off + loc[j];
    if (tid == 255) lens_out[b] = (float)(off + loc[7]);
}

// ---------------------------------------------------------------------------
// Length regulator: out[b,pos,:] = h[b, searchsorted(cum,pos,'right'), :]
// masked by pos < out_lens. One wave per output position (1 KB row copy).
// ---------------------------------------------------------------------------
__global__ __launch_bounds__(256) void gather_kernel(
    const float* __restrict__ hA, const int* __restrict__ cum,
    float* __restrict__ out)
{
    const int p    = blockIdx.x * 8 + (threadIdx.x >> 5); // 0..B*MAXOUT-1
    const int lane = threadIdx.x & 31;
    const int b    = p / MAXOUT;
    const int pos  = p % MAXOUT;
    const int* c = cum + (size_t)b * SEQ;

    int lo = 0, hi = SEQ;                 // first i with c[i] > pos
    while (lo < hi) { int mid = (lo + hi) >> 1; if (c[mid] <= pos) lo = mid + 1; else hi = mid; }
    const int idx = min(lo, SEQ - 1);
    const float scale = (pos < c[SEQ - 1]) ? 1.f : 0.f;

    const float4* src = (const float4*)(hA + ((size_t)b * SEQ + idx) * DIM);
    float4*       dst = (float4*)(out + ((size_t)b * MAXOUT + pos) * DIM);
#pragma unroll
    for (int j = 0; j < 2; ++j) {
        float4 v = src[lane + 32 * j];
        v.x *= scale; v.y *= scale; v.z *= scale; v.w *= scale;
        dst[lane + 32 * j] = v;
    }
}

// ---------------------------------------------------------------------------
extern "C" void kernel_launch(void* const* d_in, const int* in_sizes, int n_in,
                              void* d_out, int out_size, void* d_ws, size_t ws_size,
                              hipStream_t stream) {
    (void)in_sizes; (void)n_in; (void)out_size; (void)ws_size;

    const float* hidden = (const float*)d_in[0];
    const int*   dur    = (const int*)d_in[2];
    const float* const* P = (const float* const*)d_in;
    const float* embed_pitch  = (const float*)d_in[33];
    const float* embed_energy = (const float*)d_in[34];
    const float* pitch_bins   = (const float*)d_in[35];
    const float* energy_bins  = (const float*)d_in[36];

    const size_t NROW = (size_t)BATCH * SEQ;          // 65536
    float* hA   = (float*)d_ws;                       // hidden + embeddings
    float* tmp1 = hA   + NROW * DIM;
    float* tmp2 = tmp1 + NROW * DIM;
    int*   cum  = (int*)(tmp2 + NROW * DIM);
    float* wpk  = (float*)(cum + NROW);               // 6 packed weight bufs

    float* out_main = (float*)d_out;                  // (B, MAXOUT, D)
    float* out_lens = out_main + (size_t)BATCH * MAXOUT * DIM;
    float* out_ldur = out_lens + BATCH;
    float* out_pit  = out_ldur + NROW;
    float* out_en   = out_pit  + NROW;

    const dim3 blk(256);
    const dim3 gConv(NROW / 32);
    const dim3 gProj(NROW / 8);
    const dim3 gPack(WPACK / 256);

    // ---- pack all six conv weights into WMMA fragment order ----
    const float* wsrc[6] = { P[3], P[7], P[13], P[17], P[23], P[27] };
    for (int i = 0; i < 6; ++i)
        pack_w_kernel<<<gPack, blk, 0, stream>>>(wsrc[i], wpk + (size_t)i * WPACK);
    float* wp_dp1 = wpk;
    float* wp_dp2 = wpk + 1 * (size_t)WPACK;
    float* wp_pp1 = wpk + 2 * (size_t)WPACK;
    float* wp_pp2 = wpk + 3 * (size_t)WPACK;
    float* wp_ep1 = wpk + 4 * (size_t)WPACK;
    float* wp_ep2 = wpk + 5 * (size_t)WPACK;

    // ---- duration predictor (on hidden) ----
    conv_ln_kernel<<<gConv, blk, 0, stream>>>(hidden, wp_dp1, P[4], P[5], P[6], tmp1);
    conv_ln_kernel<<<gConv, blk, 0, stream>>>(tmp1,   wp_dp2, P[8], P[9], P[10], tmp2);
    proj_embed_kernel<<<gProj, blk, 0, stream>>>(tmp2, P[11], P[12],
        nullptr, nullptr, nullptr, nullptr, out_ldur, 0);

    // ---- pitch predictor (on hidden) -> hA = hidden + pitch_emb ----
    conv_ln_kernel<<<gConv, blk, 0, stream>>>(hidden, wp_pp1, P[14], P[15], P[16], tmp1);
    conv_ln_kernel<<<gConv, blk, 0, stream>>>(tmp1,   wp_pp2, P[18], P[19], P[20], tmp2);
    proj_embed_kernel<<<gProj, blk, 0, stream>>>(tmp2, P[21], P[22],
        pitch_bins, embed_pitch, hidden, hA, out_pit, 1);

    // ---- energy predictor (on hA) -> hA += energy_emb ----
    conv_ln_kernel<<<gConv, blk, 0, stream>>>(hA,   wp_ep1, P[24], P[25], P[26], tmp1);
    conv_ln_kernel<<<gConv, blk, 0, stream>>>(tmp1, wp_ep2, P[28], P[29], P[30], tmp2);
    proj_embed_kernel<<<gProj, blk, 0, stream>>>(tmp2, P[31], P[32],
        energy_bins, embed_energy, hA, hA, out_en, 2);

    // ---- length regulator ----
    cumsum_kernel<<<dim3(BATCH), blk, 0, stream>>>(dur, cum, out_lens);
    gather_kernel<<<dim3((BATCH * MAXOUT) / 8), blk, 0, stream>>>(hA, cum, out_main);
}